// GCNEncoder_2284922601980
// MI455X (gfx1250) — compile-verified
//
#include <hip/hip_runtime.h>
#include <stdint.h>

typedef __attribute__((ext_vector_type(2))) float v2f;
typedef __attribute__((ext_vector_type(8))) float v8f;

// ---------------------------------------------------------------------------
// Degree / normalization kernels
// ---------------------------------------------------------------------------
__global__ void deg_init(float* __restrict__ deg, int n) {
    int i = blockIdx.x * blockDim.x + threadIdx.x;
    if (i < n) deg[i] = 1.0f;                      // self-loop contributes 1
}

__global__ void deg_accum(const long long* __restrict__ dst,
                          float* __restrict__ deg, int e) {
    int i = blockIdx.x * blockDim.x + threadIdx.x;
    if (i < e) atomicAdd(&deg[(int)dst[i]], 1.0f);
}

__global__ void deg_rsqrt(float* __restrict__ deg, int n) {
    int i = blockIdx.x * blockDim.x + threadIdx.x;
    if (i < n) deg[i] = rsqrtf(deg[i]);            // deg >= 1 always
}

__global__ void zero_f32(float* __restrict__ p, long long n) {
    long long i = (long long)blockIdx.x * blockDim.x + threadIdx.x;
    long long stride = (long long)gridDim.x * blockDim.x;
    for (; i < n; i += stride) p[i] = 0.0f;
}

// ---------------------------------------------------------------------------
// GEMM: C[M x FOUT] = A[M x K] * B[K x FOUT]  (row-major, fp32, exact)
// One block = FOUT/16 waves, each wave computes one 16x16 tile of one
// 16-row stripe via V_WMMA_F32_16X16X4_F32 chained over K.
// A tile staged in LDS (stride 132 floats -> conflict-free ds_load_b64).
// Requires M % 16 == 0 (M = 50000 = 3125*16), K % 4 == 0, K <= 128.
// ---------------------------------------------------------------------------
#define SROW 132

template <int FOUT>
__global__ __launch_bounds__(32 * (FOUT / 16))
void gemm_wmma_f32(const float* __restrict__ A, const float* __restrict__ B,
                   float* __restrict__ C, int K) {
    __shared__ float sA[16 * SROW];
    const int tid  = threadIdx.x;
    const int row0 = blockIdx.x * 16;

    // Cooperative load of the 16 x K A-tile (float4 chunks)
    const int chunks = 16 * (K >> 2);
    for (int i = tid; i < chunks; i += 32 * (FOUT / 16)) {
        int r  = i / (K >> 2);
        int c4 = i % (K >> 2);
        float4 v = *(const float4*)(A + (size_t)(row0 + r) * K + c4 * 4);
        *(float4*)(&sA[r * SROW + c4 * 4]) = v;
    }
    __syncthreads();

    const int wave = tid >> 5;
    const int lane = tid & 31;
    const int mrow = lane & 15;            // A row within tile (lanes 0-15 / 16-31)
    const int kb   = (lane >> 4) << 1;     // K-pair select per half-wave: 0 or 2
    const int n    = wave * 16 + (lane & 15);

    v8f acc = {};
    for (int k = 0; k < K; k += 4) {
        // A operand: lane holds A[mrow][k+kb], A[mrow][k+kb+1]  (8B aligned)
        v2f a = *(const v2f*)(&sA[mrow * SROW + k + kb]);
        // B operand: lane holds B[k+kb][n], B[k+kb+1][n]
        v2f b;
        b.x = B[(size_t)(k + kb) * FOUT + n];
        b.y = B[(size_t)(k + kb + 1) * FOUT + n];
        acc = __builtin_amdgcn_wmma_f32_16x16x4_f32(
            /*neg_a=*/false, a, /*neg_b=*/false, b,
            /*c_mod=*/(short)0, acc, /*reuse_a=*/false, /*reuse_b=*/false);
    }

    // C/D layout: lanes 0-15 -> rows row0+0..7, lanes 16-31 -> rows row0+8..15
    const int rbase = row0 + ((lane >> 4) << 3);
#pragma unroll
    for (int r = 0; r < 8; ++r)
        C[(size_t)(rbase + r) * FOUT + n] = acc[r];
}

// ---------------------------------------------------------------------------
// Edge scatter: AGG[dst] += dinv[src]*dinv[dst] * H[src]   (float4 per lane)
// F/4 lanes per edge; for F=128 one wave == one edge (index loads broadcast).
// ---------------------------------------------------------------------------
__global__ void scatter_edges(const long long* __restrict__ src,
                              const long long* __restrict__ dst,
                              const float* __restrict__ dinv,
                              const float* __restrict__ H,
                              float* __restrict__ AGG,
                              int E, int F) {
    const int q = F >> 2;
    long long t = (long long)blockIdx.x * blockDim.x + threadIdx.x;
    long long total = (long long)E * q;
    if (t >= total) return;
    int e = (int)(t / q);
    int c = (int)(t % q);
    int s = (int)src[e];
    int d = (int)dst[e];
    float w = dinv[s] * dinv[d];
    float4 h = *(const float4*)(H + (size_t)s * F + c * 4);
    float* out = AGG + (size_t)d * F + c * 4;
    atomicAdd(out + 0, h.x * w);
    atomicAdd(out + 1, h.y * w);
    atomicAdd(out + 2, h.z * w);
    atomicAdd(out + 3, h.w * w);
}

// ---------------------------------------------------------------------------
// Epilogues: add self-loop term dinv[i]^2 * H[i] + bias  (+ optional ReLU)
// ---------------------------------------------------------------------------
__global__ void epilogue_relu(float* __restrict__ agg,
                              const float* __restrict__ H,
                              const float* __restrict__ dinv,
                              const float* __restrict__ bias,
                              int N, int F) {
    long long t = (long long)blockIdx.x * blockDim.x + threadIdx.x;
    if (t >= (long long)N * F) return;
    int i = (int)(t / F);
    int f = (int)(t % F);
    float di = dinv[i];
    float v = agg[t] + di * di * H[t] + bias[f];
    agg[t] = fmaxf(v, 0.0f);
}

__global__ void epilogue_bias(float* __restrict__ out,
                              const float* __restrict__ H,
                              const float* __restrict__ dinv,
                              const float* __restrict__ bias,
                              int N, int F) {
    long long t = (long long)blockIdx.x * blockDim.x + threadIdx.x;
    if (t >= (long long)N * F) return;
    int i = (int)(t / F);
    int f = (int)(t % F);
    float di = dinv[i];
    out[t] = out[t] + di * di * H[t] + bias[f];
}

// ---------------------------------------------------------------------------
// Launch
// ---------------------------------------------------------------------------
extern "C" void kernel_launch(void* const* d_in, const int* in_sizes, int n_in,
                              void* d_out, int out_size, void* d_ws, size_t ws_size,
                              hipStream_t stream) {
    const float*     x  = (const float*)d_in[0];
    const long long* ei = (const long long*)d_in[1];   // int64 edge_index [2,E]
    const float*     W1 = (const float*)d_in[2];
    const float*     b1 = (const float*)d_in[3];
    const float*     W2 = (const float*)d_in[4];
    const float*     b2 = (const float*)d_in[5];

    const int Hid  = in_sizes[3];            // 128
    const int Fout = in_sizes[5];            // 64
    const int Fin  = in_sizes[2] / Hid;      // 128
    const int N    = in_sizes[0] / Fin;      // 50000
    const int E    = in_sizes[1] / 2;        // 800000

    const long long* src = ei;
    const long long* dst = ei + E;

    // Workspace carve-up: dinv | h1 | agg1   (h2 reuses the h1 slot)
    float* dinv = (float*)d_ws;
    float* h1   = dinv + ((N + 63) & ~63);
    float* agg1 = h1 + (size_t)N * Hid;
    float* h2   = h1;                        // h1 dead after epilogue_relu
    float* out  = (float*)d_out;

    const int B = 256;
    // --- normalization -----------------------------------------------------
    deg_init <<<(N + B - 1) / B, B, 0, stream>>>(dinv, N);
    deg_accum<<<(E + B - 1) / B, B, 0, stream>>>(dst, dinv, E);
    deg_rsqrt<<<(N + B - 1) / B, B, 0, stream>>>(dinv, N);

    // --- layer 1: h1 = x @ W1 ; agg1 = scatter(norm * h1) ; relu+b1 --------
    gemm_wmma_f32<128><<<N / 16, 256, 0, stream>>>(x, W1, h1, Fin);

    long long nAgg1 = (long long)N * Hid;
    zero_f32<<<4096, B, 0, stream>>>(agg1, nAgg1);
    long long th1 = (long long)E * (Hid >> 2);
    scatter_edges<<<(unsigned)((th1 + B - 1) / B), B, 0, stream>>>(
        src, dst, dinv, h1, agg1, E, Hid);
    epilogue_relu<<<(unsigned)((nAgg1 + B - 1) / B), B, 0, stream>>>(
        agg1, h1, dinv, b1, N, Hid);

    // --- layer 2: h2 = agg1 @ W2 ; out = scatter(norm * h2) + b2 -----------
    gemm_wmma_f32<64><<<N / 16, 128, 0, stream>>>(agg1, W2, h2, Hid);

    long long nOut = (long long)N * Fout;
    zero_f32<<<4096, B, 0, stream>>>(out, nOut);
    long long th2 = (long long)E * (Fout >> 2);
    scatter_edges<<<(unsigned)((th2 + B - 1) / B), B, 0, stream>>>(
        src, dst, dinv, h2, out, E, Fout);
    epilogue_bias<<<(unsigned)((nOut + B - 1) / B), B, 0, stream>>>(
        out, h2, dinv, b2, N, Fout);
}